// GEmbedding_32555852103730
// MI455X (gfx1250) — compile-verified
//
#include <hip/hip_runtime.h>

// V=512, G=4, D=64. emb[r,:] = W[GE[r]%3,:] + b,  r = v*G+g  (2048 rows)
// out[(r*512 + v2)*64 + d] = emb[r,d]   -> 256 MB of pure streaming stores.

typedef __attribute__((ext_vector_type(2))) float v2f;
typedef __attribute__((ext_vector_type(4))) float v4f;
typedef __attribute__((ext_vector_type(8))) float v8f;

#define VV 512
#define GG 4
#define DD 64
#define ROWS_PER_BLK 16   // 16 (v,g) rows per block  -> gridDim.x = 2048/16 = 128
#define REPS_PER_BLK 16   // 16 v2 repeats per block  -> gridDim.y = 512/16  = 32

__global__ __launch_bounds__(256)
void GEmbedding_bcast_kernel(const int* __restrict__ ge,
                             const float* __restrict__ W,   // [3,64] row-major
                             const float* __restrict__ bias,// [64]
                             float* __restrict__ out) {
    __shared__ __align__(16) float lds_emb[ROWS_PER_BLK * DD]; // 4 KB tile

    const int t    = threadIdx.x;
    const int lane = t & 31;
    const int wave = t >> 5;
    const int r0   = blockIdx.x * ROWS_PER_BLK;

    // ---- Phase 1: waves 0..3 compute the 16x64 emb tile with WMMA f32 16x16x4.
    // A (16x4 f32): per row m, [onehot0, onehot1, onehot2, 1.0]  (bias trick)
    // B (4x16 f32): col tile c, rows = W[0..2][c*16+n], then bias[c*16+n]
    // => D = A*B = W[e] + b exactly, in full f32.
    if (wave < 4) {
        const int c    = wave;        // which 16-wide column tile of D (D=64 -> 4 tiles)
        const int m    = lane & 15;   // row within tile
        const int half = lane >> 4;   // 0 -> K=0,1 ; 1 -> K=2,3  (A layout)
        const unsigned e = ((unsigned)ge[r0 + m]) % 3u;

        v2f a;
        a.x = half ? (e == 2u ? 1.0f : 0.0f) : (e == 0u ? 1.0f : 0.0f);
        a.y = half ? 1.0f                    : (e == 1u ? 1.0f : 0.0f);

        const int col = c * 16 + (lane & 15); // N within full D
        v2f bb;
        bb.x = half ? W[2 * DD + col] : W[0 * DD + col];
        bb.y = half ? bias[col]       : W[1 * DD + col];

        v8f acc = {};
        acc = __builtin_amdgcn_wmma_f32_16x16x4_f32(
            /*neg_a=*/false, a, /*neg_b=*/false, bb,
            /*c_mod=*/(short)0, acc, /*reuse_a=*/false, /*reuse_b=*/false);

        // C/D layout: lanes 0-15 VGPR p = D[M=p][N=lane]; lanes 16-31 = D[M=8+p][N=lane-16]
        #pragma unroll
        for (int p = 0; p < 8; ++p) {
            const int mrow = p + 8 * half;
            lds_emb[mrow * DD + col] = acc[p];
        }
    }
    __syncthreads();

    // ---- Phase 2: broadcast stores. Each thread owns one float4 of one emb row
    // and writes it REPS_PER_BLK times (stride 256 B) -> 16 x global_store_b128 (NT).
    const int i = t >> 4;                  // row within tile: 0..15
    const int q = t & 15;                  // float4 index within 64-float row
    const v4f val = *reinterpret_cast<const v4f*>(&lds_emb[i * DD + q * 4]);

    const size_t r    = (size_t)(r0 + i);
    const size_t base = (r * (size_t)VV + (size_t)blockIdx.y * REPS_PER_BLK) * (size_t)DD
                        + (size_t)q * 4;
    float* outp = out + base;
    #pragma unroll
    for (int k = 0; k < REPS_PER_BLK; ++k) {
        __builtin_nontemporal_store(val, reinterpret_cast<v4f*>(outp + (size_t)k * DD));
    }
}

extern "C" void kernel_launch(void* const* d_in, const int* in_sizes, int n_in,
                              void* d_out, int out_size, void* d_ws, size_t ws_size,
                              hipStream_t stream) {
    (void)in_sizes; (void)n_in; (void)out_size; (void)d_ws; (void)ws_size;
    const int*   GE = (const int*)  d_in[0];   // [512,4,1] ints (0..2)
    const float* W  = (const float*)d_in[1];   // [3,64]
    const float* b  = (const float*)d_in[2];   // [64]
    float*       O  = (float*)d_out;           // [512,4,512,64]

    dim3 grid((VV * GG) / ROWS_PER_BLK, VV / REPS_PER_BLK); // 128 x 32 = 4096 blocks
    GEmbedding_bcast_kernel<<<grid, 256, 0, stream>>>(GE, W, b, O);
}